// GCN_75608604279055
// MI455X (gfx1250) — compile-verified
//
#include <hip/hip_runtime.h>

typedef __attribute__((ext_vector_type(2))) float v2f;
typedef __attribute__((ext_vector_type(8))) float v8f;

// ---------------------------------------------------------------------------
// K0: zero a float buffer
// ---------------------------------------------------------------------------
__global__ void k_zero(float* __restrict__ p, int n) {
  int i = blockIdx.x * blockDim.x + threadIdx.x;
  int s = gridDim.x * blockDim.x;
  for (; i < n; i += s) p[i] = 0.0f;
}

// ---------------------------------------------------------------------------
// K1: degree counts (float atomics; counts are small integers, exact in f32)
// ---------------------------------------------------------------------------
__global__ void k_count(const long long* __restrict__ dst, int E,
                        float* __restrict__ degcnt) {
  int i = blockIdx.x * blockDim.x + threadIdx.x;
  int s = gridDim.x * blockDim.x;
  for (; i < E; i += s) atomicAdd(&degcnt[(int)dst[i]], 1.0f);
}

// ---------------------------------------------------------------------------
// K2: per-node init. deg = cnt+1 (self loop); dinv = rsqrt(deg);
//     xw1[n,f] = x[n]*W1[f];  agg1[n,f] = xw1[n,f]/deg[n]  (self-loop term)
//     16 threads per node (one per feature).
// ---------------------------------------------------------------------------
__global__ void k_node_init(const float* __restrict__ x,
                            const float* __restrict__ W1,
                            const float* __restrict__ degcnt,
                            float* __restrict__ dinv,
                            float* __restrict__ invdeg,
                            float* __restrict__ xw1,
                            float* __restrict__ agg1, int N) {
  int tid = blockIdx.x * blockDim.x + threadIdx.x;
  if (tid >= N * 16) return;
  int n = tid >> 4;
  int f = tid & 15;
  float d  = degcnt[n] + 1.0f;
  float id = 1.0f / d;
  if (f == 0) { dinv[n] = rsqrtf(d); invdeg[n] = id; }
  float v = x[n] * W1[f];
  xw1[tid]  = v;
  agg1[tid] = v * id;
}

// ---------------------------------------------------------------------------
// K3: conv1 edge scatter. Also caches norm[e] for reuse by conv2.
//     Gathers are float4 (node rows live in L2); 16 f32 atomics per edge.
// ---------------------------------------------------------------------------
__global__ void k_scatter1(const long long* __restrict__ src,
                           const long long* __restrict__ dst,
                           const float* __restrict__ dinv,
                           const float* __restrict__ xw1,
                           float* __restrict__ agg1,
                           float* __restrict__ nrm, int E) {
  int e  = blockIdx.x * blockDim.x + threadIdx.x;
  int st = gridDim.x * blockDim.x;
  for (; e < E; e += st) {
    int s = (int)src[e];
    int d = (int)dst[e];
    float w = dinv[s] * dinv[d];
    nrm[e] = w;
    const float4* xs = (const float4*)(xw1 + (size_t)s * 16);
    float* ad = agg1 + (size_t)d * 16;
#pragma unroll
    for (int q = 0; q < 4; ++q) {
      float4 v = xs[q];
      atomicAdd(&ad[q * 4 + 0], v.x * w);
      atomicAdd(&ad[q * 4 + 1], v.y * w);
      atomicAdd(&ad[q * 4 + 2], v.z * w);
      atomicAdd(&ad[q * 4 + 3], v.w * w);
    }
  }
}

// ---------------------------------------------------------------------------
// K4: h1 = relu(agg1 + b1), in place (F = 16)
// ---------------------------------------------------------------------------
__global__ void k_bias_relu16(float* __restrict__ a, const float* __restrict__ b,
                              int total) {
  int i = blockIdx.x * blockDim.x + threadIdx.x;
  if (i >= total) return;
  a[i] = fmaxf(a[i] + b[i & 15], 0.0f);
}

// ---------------------------------------------------------------------------
// K5: xw2 = h1[100k,16] @ W2[16,32] via V_WMMA_F32_16X16X4_F32.
//     One wave per 16-row tile; 2 N-tiles (cols 0..15, 16..31); K=16 as 4
//     chunks of 4 -> 8 wmma per wave. Fuses agg2 init = xw2 * (1/deg).
//     f32 A 16x4 layout: lanes 0-15 hold K={0,1}, lanes 16-31 K={2,3};
//     B 4x16 mirrored (VGPR r: lanes 0-15 = K=r, lanes 16-31 = K=2+r).
//     C/D: VGPR r -> row r (lanes 0-15) / r+8 (lanes 16-31), col = lane%16.
// ---------------------------------------------------------------------------
__global__ void k_gemm_wmma(const float* __restrict__ h1,
                            const float* __restrict__ W2,
                            const float* __restrict__ invdeg,
                            float* __restrict__ xw2,
                            float* __restrict__ agg2, int nTiles) {
  int wave = (blockIdx.x * blockDim.x + threadIdx.x) >> 5;
  if (wave >= nTiles) return;            // uniform per wave: EXEC stays all-1s
  int lane = threadIdx.x & 31;
  int half = lane >> 4;                  // 0: K pair {0,1}; 1: K pair {2,3}
  int l16  = lane & 15;
  int mbase = wave << 4;

  v8f acc0 = {};
  v8f acc1 = {};
#pragma unroll
  for (int kc = 0; kc < 4; ++kc) {
    int k0 = kc * 4 + half * 2;
    v2f a, b0, b1;
    const float* ar = h1 + (size_t)(mbase + l16) * 16 + k0;
    a.x = ar[0];
    a.y = ar[1];
    b0.x = W2[(k0 + 0) * 32 + l16];
    b0.y = W2[(k0 + 1) * 32 + l16];
    b1.x = W2[(k0 + 0) * 32 + 16 + l16];
    b1.y = W2[(k0 + 1) * 32 + 16 + l16];
    acc0 = __builtin_amdgcn_wmma_f32_16x16x4_f32(false, a, false, b0,
                                                 (short)0, acc0, false, false);
    acc1 = __builtin_amdgcn_wmma_f32_16x16x4_f32(false, a, false, b1,
                                                 (short)0, acc1, false, false);
  }

#pragma unroll
  for (int r = 0; r < 8; ++r) {
    int row = mbase + r + half * 8;
    float id = invdeg[row];
    float v0 = acc0[r];
    float v1 = acc1[r];
    xw2[(size_t)row * 32 + l16]       = v0;
    xw2[(size_t)row * 32 + 16 + l16]  = v1;
    agg2[(size_t)row * 32 + l16]      = v0 * id;
    agg2[(size_t)row * 32 + 16 + l16] = v1 * id;
  }
}

// ---------------------------------------------------------------------------
// K6: conv2 edge scatter, reusing cached norm[e]. 32 f32 atomics per edge.
// ---------------------------------------------------------------------------
__global__ void k_scatter2(const long long* __restrict__ src,
                           const long long* __restrict__ dst,
                           const float* __restrict__ nrm,
                           const float* __restrict__ xw2,
                           float* __restrict__ agg2, int E) {
  int e  = blockIdx.x * blockDim.x + threadIdx.x;
  int st = gridDim.x * blockDim.x;
  for (; e < E; e += st) {
    int s = (int)src[e];
    int d = (int)dst[e];
    float w = nrm[e];
    const float4* xs = (const float4*)(xw2 + (size_t)s * 32);
    float* ad = agg2 + (size_t)d * 32;
#pragma unroll
    for (int q = 0; q < 8; ++q) {
      float4 v = xs[q];
      atomicAdd(&ad[q * 4 + 0], v.x * w);
      atomicAdd(&ad[q * 4 + 1], v.y * w);
      atomicAdd(&ad[q * 4 + 2], v.z * w);
      atomicAdd(&ad[q * 4 + 3], v.w * w);
    }
  }
}

// ---------------------------------------------------------------------------
// K7: mean-pool: gsum[f] += sum_n relu(agg2[n,f] + b2[f]).
//     Thread tid handles feature tid&31, nodes strided; one atomic per thread.
// ---------------------------------------------------------------------------
__global__ void k_pool(const float* __restrict__ agg2,
                       const float* __restrict__ b2,
                       float* __restrict__ gsum, int N) {
  int tid = blockIdx.x * blockDim.x + threadIdx.x;
  int f = tid & 31;
  int n = tid >> 5;
  int nstride = (gridDim.x * blockDim.x) >> 5;
  float bias = b2[f];
  float s = 0.0f;
  for (; n < N; n += nstride) s += fmaxf(agg2[(size_t)n * 32 + f] + bias, 0.0f);
  atomicAdd(&gsum[f], s);
}

// ---------------------------------------------------------------------------
// K8: MLP head, single wave.  g = gsum/N; t = relu(g@Wf1+bf1); out = t@Wf2+bf2
// ---------------------------------------------------------------------------
__global__ void k_head(const float* __restrict__ gsum,
                       const float* __restrict__ Wf1,
                       const float* __restrict__ bf1,
                       const float* __restrict__ Wf2,
                       const float* __restrict__ bf2,
                       float* __restrict__ out, float invN) {
  __shared__ float t[10];
  int l = threadIdx.x;
  if (l < 10) {
    float acc = bf1[l];
    for (int f = 0; f < 32; ++f) acc += gsum[f] * invN * Wf1[f * 10 + l];
    t[l] = fmaxf(acc, 0.0f);
  }
  __syncthreads();
  if (l < 2) {
    float acc = bf2[l];
    for (int j = 0; j < 10; ++j) acc += t[j] * Wf2[j * 2 + l];
    out[l] = acc;
  }
}

// ---------------------------------------------------------------------------
// Host-side launcher
// ---------------------------------------------------------------------------
extern "C" void kernel_launch(void* const* d_in, const int* in_sizes, int n_in,
                              void* d_out, int out_size, void* d_ws, size_t ws_size,
                              hipStream_t stream) {
  const float*     x   = (const float*)d_in[0];
  const long long* ei  = (const long long*)d_in[1];
  const float*     W1  = (const float*)d_in[2];
  const float*     b1  = (const float*)d_in[3];
  const float*     W2  = (const float*)d_in[4];
  const float*     b2  = (const float*)d_in[5];
  const float*     Wf1 = (const float*)d_in[6];
  const float*     bf1 = (const float*)d_in[7];
  const float*     Wf2 = (const float*)d_in[8];
  const float*     bf2 = (const float*)d_in[9];

  const int N = in_sizes[0];          // 100,000 nodes (x is [N,1])
  const int E = in_sizes[1] / 2;      // 3,200,000 edges
  const long long* src = ei;
  const long long* dst = ei + E;

  // Workspace carve-out (all float): ~52.4 MB total
  float* ws     = (float*)d_ws;
  float* degcnt = ws;                             // N
  float* dinv   = degcnt + N;                     // N
  float* invdeg = dinv + N;                       // N
  float* xw1    = invdeg + N;                     // N*16
  float* agg1   = xw1 + (size_t)N * 16;           // N*16  (becomes h1)
  float* xw2    = agg1 + (size_t)N * 16;          // N*32
  float* agg2   = xw2 + (size_t)N * 32;           // N*32
  float* nrm    = agg2 + (size_t)N * 32;          // E
  float* gsum   = nrm + (size_t)E;                // 32

  const int TB = 256;
  const int eb  = (E + TB - 1) / TB;
  const int nb1 = (N * 16 + TB - 1) / TB;

  // K0: zero degree counts + gsum
  k_zero<<<(N + TB - 1) / TB, TB, 0, stream>>>(degcnt, N);
  k_zero<<<1, 32, 0, stream>>>(gsum, 32);

  // K1: degree counts
  k_count<<<eb, TB, 0, stream>>>(dst, E, degcnt);

  // K2: node init (dinv/invdeg/xw1/agg1 self-loop term)
  k_node_init<<<nb1, TB, 0, stream>>>(x, W1, degcnt, dinv, invdeg, xw1, agg1, N);

  // K3: conv1 scatter (+ cache norm[e])
  k_scatter1<<<eb, TB, 0, stream>>>(src, dst, dinv, xw1, agg1, nrm, E);

  // K4: h1 = relu(agg1 + b1), in place
  k_bias_relu16<<<nb1, TB, 0, stream>>>(agg1, b1, N * 16);

  // K5: WMMA GEMM h1 @ W2 -> xw2, agg2 = xw2/deg
  int nTiles = (N + 15) / 16;                 // 6250 (N divisible by 16)
  int gemmBlocks = (nTiles * 32 + TB - 1) / TB;
  k_gemm_wmma<<<gemmBlocks, TB, 0, stream>>>(agg1, W2, invdeg, xw2, agg2, nTiles);

  // K6: conv2 scatter using cached norm
  k_scatter2<<<eb, TB, 0, stream>>>(src, dst, nrm, xw2, agg2, E);

  // K7: fused bias+relu+mean-pool accumulation
  k_pool<<<64, TB, 0, stream>>>(agg2, b2, gsum, N);

  // K8: MLP head -> d_out[2]
  k_head<<<1, 32, 0, stream>>>(gsum, Wf1, bf1, Wf2, bf2, (float*)d_out,
                               1.0f / (float)N);
}